// Quantizer_18485539242752
// MI455X (gfx1250) — compile-verified
//
#include <hip/hip_runtime.h>
#include <math.h>
#include <stdint.h>

// ---------------------------------------------------------------------------
// VQ-VAE quantizer for MI455X (gfx1250, wave32).
//   inpt     : [64, 2048, 64] fp32  -> N = 131072 rows, D = 64
//   emb_mtrx : [64, 1024]     fp32  -> K = 1024 codes
// Outputs (concatenated in d_out): q [N*D] fp32, loss [1], perplexity [1]
//
// argmin ||x - e_k||^2 == argmax ( x . e_k - 0.5*||e_k||^2 )
// Matmul on the matrix pipe via V_WMMA_F32_16X16X4_F32 (full fp32 precision).
// Codebook B-fragments are pre-packed into WMMA lane order, staged into LDS
// per-workgroup with the Tensor Data Mover (double buffered), consumed with
// ds_load_b128 (register double-buffered across WMMA quartets).
// ---------------------------------------------------------------------------

typedef __attribute__((ext_vector_type(2))) float v2f;
typedef __attribute__((ext_vector_type(4))) float v4f;
typedef __attribute__((ext_vector_type(8))) float v8f;
typedef __attribute__((ext_vector_type(4))) unsigned int v4u;
typedef __attribute__((ext_vector_type(4))) int v4i;
typedef __attribute__((ext_vector_type(8))) int v8i;

#define NROWS  (64 * 2048)    // 131072
#define DDIM   64
#define KCODES 1024

#define CT_TOTAL   (KCODES / 16)        // 64 column tiles of 16 codes
#define CT_CHUNK   8                    // column tiles staged per TDM transfer
#define NCHUNKS    (CT_TOTAL / CT_CHUNK)
#define CHUNK_FLTS (CT_CHUNK * 8 * 32 * 4)   // 8192 floats = 32 KB per buffer
#define ROWS_PER_WAVE 32                // two 16-row A tiles per wave

// ---------------------------------------------------------------------------
// Kernel 0: zero accumulators + cNorm[k] = 0.5 * sum_d E[d][k]^2
// ---------------------------------------------------------------------------
__global__ void vq_init(const float* __restrict__ E,
                        float* __restrict__ cNorm,
                        int* __restrict__ hist,
                        float* __restrict__ lossSum) {
    int k = blockIdx.x * blockDim.x + threadIdx.x;   // 0..1023
    float s = 0.0f;
    #pragma unroll 8
    for (int d = 0; d < DDIM; ++d) {
        float e = E[d * KCODES + k];
        s += e * e;
    }
    cNorm[k] = 0.5f * s;
    hist[k] = 0;
    if (k == 0) *lossSum = 0.0f;
}

// ---------------------------------------------------------------------------
// Kernel 1: pack E into WMMA B-fragment order.
// Bpack[((ct*8 + ktp)*32 + lane)*4 + {0..3}] =
//   { E[kb][col], E[kb+1][col], E[kb+4][col], E[kb+5][col] }
// with kb = ktp*8 + 2*(lane>>4), col = ct*16 + (lane&15).
// One ds_load_b128 in the argmax inner loop then feeds 4 WMMAs.
// ---------------------------------------------------------------------------
__global__ void vq_pack(const float* __restrict__ E,
                        float* __restrict__ Bpack) {
    const int tid  = blockIdx.x * blockDim.x + threadIdx.x;  // 0..16383
    const int lane = tid & 31;
    const int ktp  = (tid >> 5) & 7;
    const int ct   = tid >> 8;
    const int col  = ct * 16 + (lane & 15);
    const int kb   = ktp * 8 + 2 * (lane >> 4);
    float4 v;
    v.x = E[(size_t)(kb    ) * KCODES + col];
    v.y = E[(size_t)(kb + 1) * KCODES + col];
    v.z = E[(size_t)(kb + 4) * KCODES + col];
    v.w = E[(size_t)(kb + 5) * KCODES + col];
    ((float4*)Bpack)[tid] = v;
}

// ---------------------------------------------------------------------------
// TDM descriptor for a contiguous 32 KB (8192-dword) global -> LDS copy.
// ---------------------------------------------------------------------------
__device__ __forceinline__ void tdm_copy_chunk(const float* gsrc, uint32_t lds_byte_off) {
    const uint64_t ga = (uint64_t)(uintptr_t)gsrc;
    v4u g0;
    g0.x = 1u;                                   // count=1, user descriptor
    g0.y = lds_byte_off;                         // lds_addr
    g0.z = (uint32_t)(ga & 0xFFFFFFFFu);         // global_addr[31:0]
    g0.w = (uint32_t)((ga >> 32) & 0x01FFFFFFu)  // global_addr[56:32]
         | (2u << 30);                           // type = 2 (image)
    const uint32_t dim0 = CHUNK_FLTS;            // 8192 dwords
    v8i g1;
    g1[0] = (int)(2u << 16);                     // data_size = 4B; no flags
    g1[1] = (int)((dim0 & 0xFFFFu) << 16);       // tensor_dim0[15:0]
    g1[2] = (int)((dim0 >> 16) | (1u << 16));    // tensor_dim0[31:16], tensor_dim1=1
    g1[3] = (int)((dim0 & 0xFFFFu) << 16);       // tile_dim0 = 8192
    g1[4] = 1;                                   // tile_dim1 = 1, tile_dim2 = 0
    g1[5] = (int)dim0;                           // tensor_dim0_stride[31:0]
    g1[6] = (int)((dim0 & 0xFFFFu) << 16);       // stride0[47:32]=0, stride1[15:0]
    g1[7] = 0;                                   // stride1[47:16]
    v4i z4 = {0, 0, 0, 0};
    v8i z8 = {0, 0, 0, 0, 0, 0, 0, 0};
    __builtin_amdgcn_tensor_load_to_lds(g0, g1, z4, z4, z8, 0);
}

// ---------------------------------------------------------------------------
// Kernel 2: per-row argmax via V_WMMA_F32_16X16X4_F32.
// 256 threads (8 waves); each wave owns 32 rows (two 16-row A tiles).
// B fragments staged into LDS by the TDM (double buffered), read as b128 with
// a one-ahead register pipeline.  Accumulators start at -cNorm[col], so the
// WMMA result is directly the score x.e - 0.5*||e||^2.
// ---------------------------------------------------------------------------
__global__ void vq_argmax(const float* __restrict__ X,
                          const float* __restrict__ Bpack,
                          const float* __restrict__ cNorm,
                          int* __restrict__ idxOut) {
    __shared__ __align__(16) float ldsB[2 * CHUNK_FLTS];    // 64 KB

    const int lane = threadIdx.x & 31;
    const int wave = threadIdx.x >> 5;
    const int rowBase = (blockIdx.x * 8 + wave) * ROWS_PER_WAVE;
    const int half = lane >> 4;
    const int l16  = lane & 15;

    // A fragments: two 16x64 row tiles resident in registers.
    v2f a0[16], a1[16];
    {
        const float* xr0 = X + (size_t)(rowBase + l16) * DDIM;
        const float* xr1 = xr0 + 16 * DDIM;
        #pragma unroll
        for (int kt = 0; kt < 16; ++kt) {
            const int kb = kt * 4 + 2 * half;
            a0[kt].x = xr0[kb]; a0[kt].y = xr0[kb + 1];
            a1[kt].x = xr1[kb]; a1[kt].y = xr1[kb + 1];
        }
    }

    float bv0[8], bv1[8];
    int   bi0[8], bi1[8];
    #pragma unroll
    for (int j = 0; j < 8; ++j) {
        bv0[j] = bv1[j] = -3.4028235e38f;
        bi0[j] = bi1[j] = 0;
    }

    const uint32_t ldsBase = (uint32_t)(uintptr_t)&ldsB[0];

    // Prologue: DMA chunk 0 into buffer 0.
    if (wave == 0) {
        tdm_copy_chunk(Bpack, ldsBase);
        __builtin_amdgcn_s_wait_tensorcnt(0);
    }
    __syncthreads();

    for (int c = 0; c < NCHUNKS; ++c) {
        const int buf = c & 1;
        if (wave == 0 && (c + 1) < NCHUNKS)      // overlap next DMA with compute
            tdm_copy_chunk(Bpack + (size_t)(c + 1) * CHUNK_FLTS,
                           ldsBase + (uint32_t)((buf ^ 1) * CHUNK_FLTS * 4));
        const float* lb = &ldsB[buf * CHUNK_FLTS];

        for (int ctl = 0; ctl < CT_CHUNK; ++ctl) {
            const int col = (c * CT_CHUNK + ctl) * 16 + l16;
            const float negcn = -cNorm[col];     // fold -0.5||e||^2 into C
            v8f acc0, acc1;
            #pragma unroll
            for (int j = 0; j < 8; ++j) { acc0[j] = negcn; acc1[j] = negcn; }

            const float* tb = lb + (size_t)(ctl * 8 * 32) * 4;
            v4f cur = *(const v4f*)(tb + (size_t)lane * 4);  // ktp = 0
            #pragma unroll
            for (int ktp = 0; ktp < 8; ++ktp) {
                v4f nxt = cur;                   // one-ahead B pipeline
                if (ktp < 7)
                    nxt = *(const v4f*)(tb + (size_t)((ktp + 1) * 32 + lane) * 4);
                v2f b0; b0.x = cur.x; b0.y = cur.y;      // kt = 2*ktp
                v2f b1; b1.x = cur.z; b1.y = cur.w;      // kt = 2*ktp + 1
                acc0 = __builtin_amdgcn_wmma_f32_16x16x4_f32(false, a0[2*ktp    ], false, b0, (short)0, acc0, false, false);
                acc1 = __builtin_amdgcn_wmma_f32_16x16x4_f32(false, a1[2*ktp    ], false, b0, (short)0, acc1, false, false);
                acc0 = __builtin_amdgcn_wmma_f32_16x16x4_f32(false, a0[2*ktp + 1], false, b1, (short)0, acc0, false, false);
                acc1 = __builtin_amdgcn_wmma_f32_16x16x4_f32(false, a1[2*ktp + 1], false, b1, (short)0, acc1, false, false);
                cur = nxt;
            }
            #pragma unroll
            for (int j = 0; j < 8; ++j) {
                if (acc0[j] > bv0[j]) { bv0[j] = acc0[j]; bi0[j] = col; }
                if (acc1[j] > bv1[j]) { bv1[j] = acc1[j]; bi1[j] = col; }
            }
        }

        if (wave == 0) __builtin_amdgcn_s_wait_tensorcnt(0);
        __syncthreads();
    }

    // Reduce over the 16 lanes holding a tile's 16 columns (within each half).
    #pragma unroll
    for (int m = 1; m < 16; m <<= 1) {
        #pragma unroll
        for (int j = 0; j < 8; ++j) {
            float ov; int oi;
            ov = __shfl_xor(bv0[j], m, 16); oi = __shfl_xor(bi0[j], m, 16);
            if (ov > bv0[j] || (ov == bv0[j] && oi < bi0[j])) { bv0[j] = ov; bi0[j] = oi; }
            ov = __shfl_xor(bv1[j], m, 16); oi = __shfl_xor(bi1[j], m, 16);
            if (ov > bv1[j] || (ov == bv1[j] && oi < bi1[j])) { bv1[j] = ov; bi1[j] = oi; }
        }
    }

    if (l16 == 0) {
        #pragma unroll
        for (int j = 0; j < 8; ++j) {
            idxOut[rowBase      + j + 8 * half] = bi0[j];
            idxOut[rowBase + 16 + j + 8 * half] = bi1[j];
        }
    }
}

// ---------------------------------------------------------------------------
// Kernel 3: gather q = E[:, idx[row]], accumulate sum((q-x)^2) and histogram.
// ---------------------------------------------------------------------------
__global__ void vq_gather(const float* __restrict__ X,
                          const float* __restrict__ E,
                          const int* __restrict__ idxIn,
                          float* __restrict__ q,
                          int* __restrict__ hist,
                          float* __restrict__ lossSum) {
    __shared__ float red[256];
    const int row = blockIdx.x * blockDim.x + threadIdx.x;
    const int idx = idxIn[row];

    const float4* x4 = (const float4*)(X + (size_t)row * DDIM);
    float4*       q4 = (float4*)(q + (size_t)row * DDIM);

    float s = 0.0f;
    #pragma unroll
    for (int i = 0; i < DDIM / 4; ++i) {
        const int d = i * 4;
        float4 e;
        e.x = E[(size_t)(d + 0) * KCODES + idx];
        e.y = E[(size_t)(d + 1) * KCODES + idx];
        e.z = E[(size_t)(d + 2) * KCODES + idx];
        e.w = E[(size_t)(d + 3) * KCODES + idx];
        const float4 xv = x4[i];
        q4[i] = e;
        const float dx = e.x - xv.x, dy = e.y - xv.y;
        const float dz = e.z - xv.z, dw = e.w - xv.w;
        s += dx * dx + dy * dy + dz * dz + dw * dw;
    }
    atomicAdd(&hist[idx], 1);

    red[threadIdx.x] = s;
    __syncthreads();
    for (int off = 128; off > 0; off >>= 1) {
        if (threadIdx.x < off) red[threadIdx.x] += red[threadIdx.x + off];
        __syncthreads();
    }
    if (threadIdx.x == 0) atomicAdd(lossSum, red[0]);
}

// ---------------------------------------------------------------------------
// Kernel 4: finalize loss and perplexity scalars.
// ---------------------------------------------------------------------------
__global__ void vq_finalize(const int* __restrict__ hist,
                            const float* __restrict__ lossSum,
                            float* __restrict__ outScalars) {
    __shared__ float red[256];
    float s = 0.0f;
    for (int k = threadIdx.x; k < KCODES; k += 256) {
        const float p = (float)hist[k] * (1.0f / (float)NROWS);
        s += p * logf(p + 1e-10f);
    }
    red[threadIdx.x] = s;
    __syncthreads();
    for (int off = 128; off > 0; off >>= 1) {
        if (threadIdx.x < off) red[threadIdx.x] += red[threadIdx.x + off];
        __syncthreads();
    }
    if (threadIdx.x == 0) {
        // l_commitment == l_codebook numerically -> loss = 2 * mean((q-x)^2)
        outScalars[0] = 2.0f * (*lossSum) / (float)((size_t)NROWS * DDIM);
        outScalars[1] = expf(-red[0]);
    }
}

// ---------------------------------------------------------------------------
extern "C" void kernel_launch(void* const* d_in, const int* in_sizes, int n_in,
                              void* d_out, int out_size, void* d_ws, size_t ws_size,
                              hipStream_t stream) {
    const float* X = (const float*)d_in[0];   // [131072, 64]
    const float* E = (const float*)d_in[1];   // [64, 1024]
    float* q = (float*)d_out;                 // q then {loss, perplexity}
    float* outScalars = q + (size_t)NROWS * DDIM;

    // Workspace layout (all 16-byte aligned regions first):
    float* Bpack   = (float*)d_ws;                          // [65536] = 256 KB
    float* cNorm   = Bpack + 65536;                         // [1024]
    int*   idx     = (int*)(cNorm + KCODES);                // [131072]
    int*   hist    = idx + NROWS;                           // [1024]
    float* lossSum = (float*)(hist + KCODES);               // [1]

    vq_init<<<KCODES / 256, 256, 0, stream>>>(E, cNorm, hist, lossSum);
    vq_pack<<<64, 256, 0, stream>>>(E, Bpack);

    // 8 waves per block, 32 rows per wave -> 256 rows per block.
    vq_argmax<<<NROWS / 256, 256, 0, stream>>>(X, Bpack, cNorm, idx);

    vq_gather<<<NROWS / 256, 256, 0, stream>>>(X, E, idx, q, hist, lossSum);

    vq_finalize<<<1, 256, 0, stream>>>(hist, lossSum, outScalars);
}